// GCN_31834297598206
// MI455X (gfx1250) — compile-verified
//
#include <hip/hip_runtime.h>

#define NNODES 100000
#define NEDGES 1600000
#define NGRAPH 128

typedef float v2f __attribute__((ext_vector_type(2)));
typedef float v8f __attribute__((ext_vector_type(8)));

// ---------------------------------------------------------------- utilities
__global__ void zero_kernel(float* __restrict__ p, int n) {
    int i = blockIdx.x * blockDim.x + threadIdx.x;
    if (i < n) p[i] = 0.0f;
}

// deg_out[src[e]] += 1 ; deg_in[dst[e]] += 1  (f32 atomics are exact here)
__global__ void degree_kernel(const int* __restrict__ src, const int* __restrict__ dst,
                              float* __restrict__ deg_out, float* __restrict__ deg_in, int e) {
    int i = blockIdx.x * blockDim.x + threadIdx.x;
    if (i < e) {
        atomicAdd(&deg_out[src[i]], 1.0f);
        atomicAdd(&deg_in[dst[i]], 1.0f);
    }
}

// p[i] = rsqrt(max(p[i], 1))   (applied to both degree arrays, contiguous)
__global__ void rsqrt_kernel(float* __restrict__ p, int n) {
    int i = blockIdx.x * blockDim.x + threadIdx.x;
    if (i < n) p[i] = rsqrtf(fmaxf(p[i], 1.0f));
}

// ------------------------------------------------- WMMA fp32 GEMM (16-row/wave)
// h[r, 0:OUT] = (x[r, 0:IN] * norm[r]) @ W[IN, OUT]   (W row-major)
// Uses V_WMMA_F32_16X16X4_F32. Per ISA layouts:
//   A 16x4 : lanes 0-15 hold K=kk,kk+1 ; lanes 16-31 hold K=kk+2,kk+3
//   B 4x16 : mirrored over N (same K split per lane-half)
//   C/D    : VGPR v -> M = v + 8*(lane>>4), N = lane&15
template <int IN, int OUT>
__global__ void gcn_gemm_wmma(const float* __restrict__ x,
                              const float* __restrict__ norm,
                              const float* __restrict__ W,
                              float* __restrict__ h, int nrows) {
    const int lane = threadIdx.x & 31;
    const int wave = threadIdx.x >> 5;
    const int tile = blockIdx.x * (blockDim.x >> 5) + wave;  // wave-uniform
    if (tile * 16 >= nrows) return;                          // uniform exit: EXEC stays all-1s

    const int r0   = tile * 16;
    const int half = lane >> 4;   // 0 or 1 (selects K pair)
    const int l16  = lane & 15;
    const int arow = r0 + l16;
    const float nrm = norm[arow];

    constexpr int NT = OUT / 16;  // N-tiles of 16 columns
    v8f acc[NT];
    const v8f vzero = {0.f, 0.f, 0.f, 0.f, 0.f, 0.f, 0.f, 0.f};
#pragma unroll
    for (int t = 0; t < NT; ++t) acc[t] = vzero;

#pragma unroll 4
    for (int kk = 0; kk < IN; kk += 4) {
        const int ka = kk + 2 * half;
        v2f a;
        a.x = x[arow * IN + ka + 0] * nrm;
        a.y = x[arow * IN + ka + 1] * nrm;
#pragma unroll
        for (int t = 0; t < NT; ++t) {
            const int col = t * 16 + l16;
            v2f b;
            b.x = W[(ka + 0) * OUT + col];
            b.y = W[(ka + 1) * OUT + col];
            acc[t] = __builtin_amdgcn_wmma_f32_16x16x4_f32(
                false, a, false, b, (short)0, acc[t], false, false);
        }
    }
#pragma unroll
    for (int t = 0; t < NT; ++t) {
        const int col = t * 16 + l16;
#pragma unroll
        for (int v = 0; v < 8; ++v) {
            h[(r0 + v + 8 * half) * OUT + col] = acc[t][v];
        }
    }
}

// ------------------------------------- edge gather + scatter-add (L2 resident)
// One thread per (edge, float4-chunk): agg[dst[e], q:q+4] += h[src[e], q:q+4]
template <int F>
__global__ void edge_agg(const int* __restrict__ src, const int* __restrict__ dst,
                         const float* __restrict__ h, float* __restrict__ agg, int e) {
    constexpr int CHUNKS = F / 4;
    int t = blockIdx.x * blockDim.x + threadIdx.x;
    int ei = t / CHUNKS;
    if (ei >= e) return;
    int q = (t % CHUNKS) * 4;
    const float4 v = *(const float4*)(h + src[ei] * F + q);
    float* p = agg + dst[ei] * F + q;
    atomicAdd(p + 0, v.x);
    atomicAdd(p + 1, v.y);
    atomicAdd(p + 2, v.z);
    atomicAdd(p + 3, v.w);
}

// h[i,j] = relu(h[i,j] * norm_in[i] + b[j])   (F = 64, in place)
__global__ void relu_bias_kernel(float* __restrict__ h, const float* __restrict__ norm_in,
                                 const float* __restrict__ b, int n) {
    int i = blockIdx.x * blockDim.x + threadIdx.x;
    if (i < n) {
        float v = fmaf(h[i], norm_in[i >> 6], b[i & 63]);
        h[i] = fmaxf(v, 0.0f);
    }
}

// per-node: out[g, :] += agg2[i, :]*norm_in[i] + b2 ; counts[g] += 1
__global__ void pool_kernel(const float* __restrict__ agg2, const float* __restrict__ norm_in,
                            const float* __restrict__ b2, const int* __restrict__ gid,
                            float* __restrict__ out, float* __restrict__ counts, int n) {
    int i = blockIdx.x * blockDim.x + threadIdx.x;
    if (i >= n) return;
    const int g = gid[i];
    const float ni = norm_in[i];
#pragma unroll
    for (int j = 0; j < 16; ++j) {
        float v = fmaf(agg2[i * 16 + j], ni, b2[j]);
        atomicAdd(&out[g * 16 + j], v);
    }
    atomicAdd(&counts[g], 1.0f);
}

__global__ void finalize_kernel(float* __restrict__ out, const float* __restrict__ counts, int n) {
    int i = blockIdx.x * blockDim.x + threadIdx.x;
    if (i < n) out[i] = out[i] / fmaxf(counts[i >> 4], 1.0f);
}

// ---------------------------------------------------------------------------
extern "C" void kernel_launch(void* const* d_in, const int* in_sizes, int n_in,
                              void* d_out, int out_size, void* d_ws, size_t ws_size,
                              hipStream_t stream) {
    const float* x   = (const float*)d_in[0];
    const float* W1  = (const float*)d_in[1];
    const float* b1  = (const float*)d_in[2];
    const float* W2  = (const float*)d_in[3];
    const float* b2  = (const float*)d_in[4];
    const int*   src = (const int*)d_in[5];
    const int*   dst = (const int*)d_in[6];
    const int*   gid = (const int*)d_in[7];
    float* out = (float*)d_out;

    // Workspace layout (floats). Zeroed region first, then overwrite-only region.
    float* ws       = (float*)d_ws;
    float* norm_out = ws;                  // N   (degrees, then rsqrt in place)
    float* norm_in  = ws + NNODES;         // N
    float* agg1     = ws + 2 * NNODES;     // 64N (scatter target; relu in place)
    float* agg2     = agg1 + 64 * NNODES;  // 16N
    float* counts   = agg2 + 16 * NNODES;  // G
    float* h1       = counts + NGRAPH;     // 64N (GEMM1 out; reused for GEMM2 out)

    const int zn = 82 * NNODES + NGRAPH;   // everything that must start at 0
    zero_kernel<<<(zn + 255) / 256, 256, 0, stream>>>(ws, zn);
    zero_kernel<<<(out_size + 255) / 256, 256, 0, stream>>>(out, out_size);

    // degrees -> norms
    degree_kernel<<<(NEDGES + 255) / 256, 256, 0, stream>>>(src, dst, norm_out, norm_in, NEDGES);
    rsqrt_kernel<<<(2 * NNODES + 255) / 256, 256, 0, stream>>>(ws, 2 * NNODES);

    const int mtiles = NNODES / 16;        // 6250 exactly
    const int gemm_blocks = (mtiles + 3) / 4;  // 4 waves (128 thr) per block

    // layer 1: h1 = (x * norm_out) @ W1 ; agg1 = scatter_add(h1[src] -> dst)
    gcn_gemm_wmma<128, 64><<<gemm_blocks, 128, 0, stream>>>(x, norm_out, W1, h1, NNODES);
    edge_agg<64><<<(NEDGES * 16 + 255) / 256, 256, 0, stream>>>(src, dst, h1, agg1, NEDGES);
    relu_bias_kernel<<<(64 * NNODES + 255) / 256, 256, 0, stream>>>(agg1, norm_in, b1, 64 * NNODES);

    // layer 2: h2 (reuses h1 buffer) = (relu1 * norm_out) @ W2 ; agg2 = scatter_add
    gcn_gemm_wmma<64, 16><<<gemm_blocks, 128, 0, stream>>>(agg1, norm_out, W2, h1, NNODES);
    edge_agg<16><<<(NEDGES * 4 + 255) / 256, 256, 0, stream>>>(src, dst, h1, agg2, NEDGES);

    // mean pooling per graph
    pool_kernel<<<(NNODES + 255) / 256, 256, 0, stream>>>(agg2, norm_in, b2, gid, out, counts, NNODES);
    finalize_kernel<<<(out_size + 255) / 256, 256, 0, stream>>>(out, counts, out_size);
}